// HistByProfMultiObj_42425686950254
// MI455X (gfx1250) — compile-verified
//
#include <hip/hip_runtime.h>
#include <cstdint>
#include <cstddef>

// ---------------- configuration ----------------
#define THREADS   256          // 8 waves (wave32)
#define TILE      1024         // floats per staged tile (4 KB)
#define NTILES    4            // 4096 pixels / 1024
#define PIX       4096         // h*w
#define NCH       128
#define NEDGES    8
#define NBINS     9

// ---------------- CDNA5 async global->LDS path ----------------
#if defined(__has_builtin)
#  if __has_builtin(__builtin_amdgcn_global_load_async_to_lds_b128)
#    define HAS_ASYNC_LDS 1
#  else
#    define HAS_ASYNC_LDS 0
#  endif
#else
#  define HAS_ASYNC_LDS 0
#endif

// b128 payload type per the builtin's signature: pointer to 16-byte int vector
typedef int v4i __attribute__((vector_size(16)));
typedef __attribute__((address_space(1))) v4i gv4i;   // global
typedef __attribute__((address_space(3))) v4i lv4i;   // LDS

__device__ __forceinline__ void copy16_to_lds(const float* g, float* l) {
#if HAS_ASYNC_LDS
  __builtin_amdgcn_global_load_async_to_lds_b128((gv4i*)g, (lv4i*)l,
                                                 /*offset=*/0, /*cpol=*/0);
#else
  // synchronous fallback (still correct with the surrounding barriers)
  *(float4*)l = *(const float4*)g;
#endif
}

#if HAS_ASYNC_LDS
#  if __has_builtin(__builtin_amdgcn_s_wait_asynccnt)
#    define WAIT_ASYNC(n) __builtin_amdgcn_s_wait_asynccnt(n)
#  else
#    define WAIT_ASYNC(n) asm volatile("s_wait_asynccnt " #n ::: "memory")
#  endif
#else
#  define WAIT_ASYNC(n) ((void)0)
#endif

// ---------------- per-tile compute ----------------
// Each thread owns 4 contiguous floats of the staged tile.
__device__ __forceinline__ void compute_tile(const float* lp,
                                             const float* a, const float* b,
                                             float negS, float sb, float* acc) {
  float4 v = *(const float4*)lp;          // ds_load_b128
  float xs[4] = {v.x, v.y, v.z, v.w};
#pragma unroll
  for (int k = 0; k < 4; ++k) {
    const float xv = xs[k];
#pragma unroll
    for (int j = 0; j < NEDGES; ++j) {
      // exp(-0.5*z^2) == exp2(-(a'*x + b')^2)  (constants pre-folded)
      const float zp = __builtin_fmaf(xv, a[j], b[j]);
      acc[j] += __builtin_amdgcn_exp2f(-(zp * zp));
    }
    // sigmoid(20*(x - e7)) = 1 / (1 + exp2(-20*log2(e)*(x - e7)))
    const float t  = __builtin_fmaf(xv, negS, sb);
    const float ev = __builtin_amdgcn_exp2f(t);
    acc[NBINS - 1] += __builtin_amdgcn_rcpf(1.0f + ev);
  }
}

// ---------------- kernel: one block per (bt, c) ----------------
__global__ __launch_bounds__(THREADS)
void hist_by_prof_kernel(const float* __restrict__ x,
                         const float* __restrict__ edges,
                         float* __restrict__ out) {
  __shared__ float buf[2][TILE];
  __shared__ float red[THREADS / 32][NBINS];

  const int tid = threadIdx.x;
  const int blk = blockIdx.x;                 // = bt*NCH + c
  const int c   = blk & (NCH - 1);

  // ---- per-channel constants (uniform across block; scalarized by compiler)
  const float* e = edges + c * NEDGES;
  const float S2   = 0.8493218002880191f;     // sqrt(0.5*log2(e))
  const float SLOG = 28.853900817779268f;     // 20*log2(e)

  float a[NEDGES], b[NEDGES];
  {
    // bin 0: mu = e0, sigma = (e0 - e1)/3
    const float inv0 = S2 * 3.0f / (e[0] - e[1]);
    a[0] = inv0;
    b[0] = -e[0] * inv0;
    float prev = e[0];
#pragma unroll
    for (int j = 1; j < NEDGES; ++j) {
      const float ej  = e[j];
      const float mu  = 0.5f * (prev + ej);
      const float inv = S2 * 3.0f / (prev - ej);
      a[j] = inv;
      b[j] = -mu * inv;
      prev = ej;
    }
  }
  const float negS = -SLOG;
  const float sb   = SLOG * e[NEDGES - 1];

  float acc[NBINS];
#pragma unroll
  for (int j = 0; j < NBINS; ++j) acc[j] = 0.0f;

  // ---- double-buffered async pipeline over 4 tiles
  const float* xg = x + (size_t)blk * PIX;
  const int    lo = tid * 4;                  // this thread's 16B slot

  copy16_to_lds(xg + 0 * TILE + lo, &buf[0][lo]);   // tile0 -> buf0
  copy16_to_lds(xg + 1 * TILE + lo, &buf[1][lo]);   // tile1 -> buf1

  WAIT_ASYNC(1);  __syncthreads();                  // tile0 resident
  compute_tile(&buf[0][lo], a, b, negS, sb, acc);
  __syncthreads();
  copy16_to_lds(xg + 2 * TILE + lo, &buf[0][lo]);   // tile2 -> buf0

  WAIT_ASYNC(1);  __syncthreads();                  // tile1 resident
  compute_tile(&buf[1][lo], a, b, negS, sb, acc);
  __syncthreads();
  copy16_to_lds(xg + 3 * TILE + lo, &buf[1][lo]);   // tile3 -> buf1

  WAIT_ASYNC(1);  __syncthreads();                  // tile2 resident
  compute_tile(&buf[0][lo], a, b, negS, sb, acc);
  __syncthreads();

  WAIT_ASYNC(0);  __syncthreads();                  // tile3 resident
  compute_tile(&buf[1][lo], a, b, negS, sb, acc);

  // ---- wave32 butterfly reduction, then cross-wave via LDS
#pragma unroll
  for (int j = 0; j < NBINS; ++j) {
    float v = acc[j];
#pragma unroll
    for (int m = 16; m >= 1; m >>= 1)
      v += __shfl_xor(v, m, 32);
    acc[j] = v;
  }
  const int wave = tid >> 5;
  const int lane = tid & 31;
  if (lane < NBINS) red[wave][lane] = acc[lane];
  __syncthreads();

  if (tid < NBINS) {
    float s = 0.0f;
#pragma unroll
    for (int w = 0; w < THREADS / 32; ++w) s += red[w][tid];
    out[(size_t)blk * NBINS + tid] = s;
  }
}

// ---------------- launcher ----------------
extern "C" void kernel_launch(void* const* d_in, const int* in_sizes, int n_in,
                              void* d_out, int out_size, void* d_ws, size_t ws_size,
                              hipStream_t stream) {
  const float* x     = (const float*)d_in[0];   // [64,128,64,64] f32
  const float* edges = (const float*)d_in[1];   // [128,8] f32
  float* out         = (float*)d_out;           // [64,128,9] f32

  const int nblocks = in_sizes[0] / PIX;        // 64*128 = 8192
  hipLaunchKernelGGL(hist_by_prof_kernel, dim3(nblocks), dim3(THREADS), 0, stream,
                     x, edges, out);
}